// AMPLayer_24799141167508
// MI455X (gfx1250) — compile-verified
//
#include <hip/hip_runtime.h>

#define N_NODES 50000
#define K_NB    32
#define FN      128
#define FE      64
#define TILE_M  16
#define SN_STR  132   // padded row stride (floats) for 16-row tiles: (4*m + c) % 64 unique banks

typedef __attribute__((ext_vector_type(2))) float v2f;
typedef __attribute__((ext_vector_type(4))) float v4f;
typedef __attribute__((ext_vector_type(8))) float v8f;

// ---------------------------------------------------------------------------
// Kernel 0: W = wq @ wk^T   (W[i][j] = sum_e wq[i][e] * wk[j][e]),  W: [FN][FE]
// Tiny (8192 dots of length 64); runs once before the main kernel.
// ---------------------------------------------------------------------------
__global__ void amp_precompute_w(const float* __restrict__ wq,
                                 const float* __restrict__ wk,
                                 float* __restrict__ W) {
    int o = blockIdx.x * blockDim.x + threadIdx.x;
    if (o >= FN * FE) return;
    int i = o >> 6;      // row in wq   (FE == 64)
    int j = o & 63;      // row in wk
    const float* a = wq + i * FE;
    const float* b = wk + j * FE;
    float s = 0.f;
#pragma unroll
    for (int e = 0; e < FE; ++e) s = fmaf(a[e], b[e], s);
    W[o] = s;
}

// ---------------------------------------------------------------------------
// Main fused kernel: one block = 16 nodes, 256 threads (8 wave32).
// ---------------------------------------------------------------------------
__global__ __launch_bounds__(256)
void amp_main(const float* __restrict__ nodes,      // [N][FN]   (L2-resident, reused)
              const int*   __restrict__ nlist,      // [N][K]
              const float* __restrict__ edges,      // [N][K][FE] (streamed once -> NT)
              const float* __restrict__ inv_degree, // [N]
              const float* __restrict__ W,          // [FN][FE] = wq @ wk^T
              const float* __restrict__ wv,         // [FN][FN]
              float*       __restrict__ out) {      // [N][FN]   (write once -> NT)
    __shared__ float sW[FN * FE];             // 32 KB
    __shared__ float sNodes[TILE_M * SN_STR]; // 8.25 KB (padded)
    __shared__ float sP[TILE_M * FE];         // 4 KB
    __shared__ float sQ[TILE_M * K_NB];       // 2 KB (logits, then softmax weights)
    __shared__ float sAgg[TILE_M * SN_STR];   // 8.25 KB (padded)

    const int tid  = threadIdx.x;
    const int lane = tid & 31;
    const int wave = tid >> 5;
    const int node_base = blockIdx.x * TILE_M;

    // ---- Phase 0: stage W and the 16-node feature tile into LDS ----
    for (int i = tid; i < FN * FE; i += 256) sW[i] = W[i];
    for (int i = tid; i < TILE_M * FN; i += 256) {
        int m = i >> 7, f = i & 127;
        sNodes[m * SN_STR + f] = nodes[(size_t)(node_base + m) * FN + f];
    }
    __syncthreads();

    // ---- Phase 1: sP[16x64] = sNodes[16x128] @ sW[128x64]  (waves 0..3, WMMA f32) ----
    if (wave < 4) {
        const int ntile = wave;                 // output cols ntile*16 .. +15
        const int m     = lane & 15;            // A row / B col within tile
        const int koff  = (lane >> 4) << 1;     // 0 for lanes 0-15, 2 for lanes 16-31
        v8f acc = {};
#pragma unroll
        for (int k0 = 0; k0 < FN; k0 += 4) {
            v2f a, b;
            a.x = sNodes[m * SN_STR + k0 + koff];
            a.y = sNodes[m * SN_STR + k0 + koff + 1];
            b.x = sW[(k0 + koff)     * FE + ntile * 16 + m];
            b.y = sW[(k0 + koff + 1) * FE + ntile * 16 + m];
            acc = __builtin_amdgcn_wmma_f32_16x16x4_f32(false, a, false, b,
                                                        (short)0, acc, false, false);
        }
        const int rbase = (lane >> 4) << 3;     // lanes 16-31 hold rows M+8
#pragma unroll
        for (int i = 0; i < 8; ++i)
            sP[(rbase + i) * FE + ntile * 16 + m] = acc[i];
    }
    __syncthreads();

    // ---- Phase 2: logits  sQ[node][k] = invdeg * (edges[n,k] . p[n]) ----
    // 16 lanes cooperate per (node,k) pair; fully-coalesced float4 edge stream.
    // Edges are read exactly once across the whole dispatch (410 MB > L2), so
    // load them non-temporally to keep nodes/wv/nlist resident in the 192MB L2.
    {
        const int grp = tid >> 4;   // 0..15
        const int li  = tid & 15;
        const v4f* e4 = reinterpret_cast<const v4f*>(edges)
                      + (size_t)node_base * (K_NB * FE / 4);
#pragma unroll 4
        for (int iter = 0; iter < 32; ++iter) {
            int pair = iter * 16 + grp;         // 0..511
            int node = pair >> 5;
            int k    = pair & 31;
            v4f ev = __builtin_nontemporal_load(
                         e4 + node * (K_NB * FE / 4) + k * (FE / 4) + li);
            const v4f pv = *reinterpret_cast<const v4f*>(&sP[node * FE + li * 4]);
            float part = ev.x * pv.x + ev.y * pv.y + ev.z * pv.z + ev.w * pv.w;
            part += __shfl_xor(part, 8, 16);
            part += __shfl_xor(part, 4, 16);
            part += __shfl_xor(part, 2, 16);
            part += __shfl_xor(part, 1, 16);
            if (li == 0)
                sQ[node * K_NB + k] = part * inv_degree[node_base + node];
        }
    }
    __syncthreads();

    // ---- Phase 3: softmax over k (16-lane group per node, 2 entries/lane) ----
    {
        const int node = tid >> 4;
        const int li   = tid & 15;
        float q0 = sQ[node * K_NB + li];
        float q1 = sQ[node * K_NB + 16 + li];
        float mx = fmaxf(q0, q1);
        mx = fmaxf(mx, __shfl_xor(mx, 8, 16));
        mx = fmaxf(mx, __shfl_xor(mx, 4, 16));
        mx = fmaxf(mx, __shfl_xor(mx, 2, 16));
        mx = fmaxf(mx, __shfl_xor(mx, 1, 16));
        float e0 = __expf(q0 - mx), e1 = __expf(q1 - mx);
        float s = e0 + e1;
        s += __shfl_xor(s, 8, 16);
        s += __shfl_xor(s, 4, 16);
        s += __shfl_xor(s, 2, 16);
        s += __shfl_xor(s, 1, 16);
        float rs = 1.0f / s;
        sQ[node * K_NB + li]      = e0 * rs;   // disjoint slots per thread: race-free
        sQ[node * K_NB + 16 + li] = e1 * rs;
    }
    __syncthreads();

    // ---- Phase 4: sAgg[node] = sum_k b[node,k] * nodes[nlist[node,k]] ----
    // One wave per 2 nodes; full-wave float4 row loads (nodes stays L2-resident
    // thanks to the NT edges stream above).
    {
        const v4f* n4 = reinterpret_cast<const v4f*>(nodes);
#pragma unroll
        for (int s = 0; s < 2; ++s) {
            const int node  = wave * 2 + s;
            const int gnode = node_base + node;
            v4f acc = {0.f, 0.f, 0.f, 0.f};
            for (int k = 0; k < K_NB; ++k) {
                int   idx = nlist[(size_t)gnode * K_NB + k];
                float bw  = sQ[node * K_NB + k];
                v4f row = n4[(size_t)idx * (FN / 4) + lane];
                acc.x = fmaf(bw, row.x, acc.x);
                acc.y = fmaf(bw, row.y, acc.y);
                acc.z = fmaf(bw, row.z, acc.z);
                acc.w = fmaf(bw, row.w, acc.w);
            }
            *reinterpret_cast<v4f*>(&sAgg[node * SN_STR + lane * 4]) = acc;
        }
    }
    __syncthreads();

    // ---- Phase 5: out tile[16x128] = sAgg[16x128] @ wv[128x128]  (8 waves, WMMA f32) ----
    {
        const int ntile = wave;                 // 0..7 -> output cols ntile*16..+15
        const int m     = lane & 15;
        const int koff  = (lane >> 4) << 1;
        const int col   = ntile * 16 + m;
        v8f acc = {};
#pragma unroll
        for (int k0 = 0; k0 < FN; k0 += 4) {
            v2f a, b;
            a.x = sAgg[m * SN_STR + k0 + koff];
            a.y = sAgg[m * SN_STR + k0 + koff + 1];
            b.x = wv[(size_t)(k0 + koff)     * FN + col];
            b.y = wv[(size_t)(k0 + koff + 1) * FN + col];
            acc = __builtin_amdgcn_wmma_f32_16x16x4_f32(false, a, false, b,
                                                        (short)0, acc, false, false);
        }
        const int rbase = (lane >> 4) << 3;
#pragma unroll
        for (int i = 0; i < 8; ++i)
            __builtin_nontemporal_store(
                acc[i], &out[(size_t)(node_base + rbase + i) * FN + col]);
    }
}

// ---------------------------------------------------------------------------
extern "C" void kernel_launch(void* const* d_in, const int* in_sizes, int n_in,
                              void* d_out, int out_size, void* d_ws, size_t ws_size,
                              hipStream_t stream) {
    (void)in_sizes; (void)n_in; (void)out_size; (void)ws_size;
    const float* nodes      = (const float*)d_in[0];
    const int*   nlist      = (const int*)  d_in[1];
    const float* edges      = (const float*)d_in[2];
    const float* inv_degree = (const float*)d_in[3];
    const float* wq         = (const float*)d_in[4];
    const float* wk         = (const float*)d_in[5];
    const float* wv         = (const float*)d_in[6];
    float* W   = (float*)d_ws;     // [FN][FE] = 32 KB scratch
    float* out = (float*)d_out;

    amp_precompute_w<<<(FN * FE + 255) / 256, 256, 0, stream>>>(wq, wk, W);
    amp_main<<<N_NODES / TILE_M, 256, 0, stream>>>(nodes, nlist, edges,
                                                   inv_degree, W, wv, out);
}